// KLDLossPlus_18769007083962
// MI455X (gfx1250) — compile-verified
//
#include <hip/hip_runtime.h>
#include <hip/hip_bf16.h>

// ---------------------------------------------------------------------------
// Problem constants (from setup_inputs)
// ---------------------------------------------------------------------------
#define BB 16          // batch
#define NP0 100        // protos per scale
#define NCLS 20        // classes per scale
#define NTILE 7        // ceil(100/15) row-tiles of 15 protos (3 classes) each
#define HW 512         // label height/width

typedef __attribute__((ext_vector_type(2))) float v2f;
typedef __attribute__((ext_vector_type(8))) float v8f;

// ---------------------------------------------------------------------------
// 0) zero counts + T accumulators
// ---------------------------------------------------------------------------
__global__ void kld_zero_kernel(int* p, int nwords) {
    int i = blockIdx.x * blockDim.x + threadIdx.x;
    if (i < nwords) p[i] = 0;
}

// ---------------------------------------------------------------------------
// 1) nearest-downsample labels (shift -1) + per-(b,class) pixel histogram
//    grid = (B, 8 chunks, 2 scales), block = 256
// ---------------------------------------------------------------------------
__global__ void kld_labels_kernel(const int* __restrict__ labels,
                                  int* __restrict__ labFlat0,
                                  int* __restrict__ labFlat1,
                                  int* __restrict__ counts) {
    const int b = blockIdx.x, ch = blockIdx.y, s = blockIdx.z;
    const int Ws   = s ? 64 : 128;
    const int N    = s ? 4096 : 16384;
    const int step = s ? 8 : 4;               // (dst * 512) / Hs
    int* labFlat = s ? labFlat1 : labFlat0;

    __shared__ int hist[32];
    if (threadIdx.x < 32) hist[threadIdx.x] = 0;
    __syncthreads();

    const int chunk = N >> 3;
    const int nEnd = (ch + 1) * chunk;
    for (int n = ch * chunk + threadIdx.x; n < nEnd; n += 256) {
        int i = n / Ws, j = n - i * Ws;
        int v = labels[((size_t)b * HW + i * step) * HW + j * step] - 1;
        labFlat[(size_t)b * N + n] = v;
        if (v >= 0) atomicAdd(&hist[v & 31], 1);
    }
    __syncthreads();
    if (threadIdx.x < NCLS && hist[threadIdx.x] != 0)
        atomicAdd(&counts[(s * BB + b) * NCLS + threadIdx.x], hist[threadIdx.x]);
}

// ---------------------------------------------------------------------------
// 2) Streaming WMMA accumulation of per-tile Gram matrices.
//    D[a][c] += exp(d_a[n]) * d_c[n]  over masked pixels (mask baked into A)
//    D[a][15] = Z_a  (B column 15 forced to 1.0)
//    grid = (NSPLIT, NTILE, B), block = 256 (8 waves, each a pixel stripe)
//    Templated on N/NSPLIT so trip counts are constexpr (no remainder path).
// ---------------------------------------------------------------------------
template <int N, int NSPLIT>
__global__ void kld_accum_kernel(const float* __restrict__ dist,
                                 const int* __restrict__ labFlat,
                                 const int* __restrict__ protoCls,
                                 float* __restrict__ T) {
    const int np = blockIdx.x;       // pixel partition
    const int t  = blockIdx.y;       // 15-proto tile
    const int b  = blockIdx.z;

    const int lane = threadIdx.x & 31;
    const int w    = threadIdx.x >> 5;
    const int r    = lane & 15;              // matrix row (A) / col (B)
    const int kk   = (lane >> 4) << 1;       // K offset: 0 or 2

    const int p = t * 15 + r;
    const bool rowValid = (r < 15) && (p < NP0);
    const int pc  = rowValid ? p : 0;
    const int cls = rowValid ? protoCls[p] : -2;   // never matches a label
    const bool onesCol = (r == 15);

    constexpr int segLen = N / (NSPLIT * 8);
    constexpr int iters  = segLen >> 2;            // WMMAs per wave

    const int nbase = (np * 8 + w) * segLen;
    const float* __restrict__ drow = dist + ((size_t)b * NP0 + pc) * (size_t)N + nbase;
    const int*   __restrict__ lrow = labFlat + (size_t)b * (size_t)N + nbase;

    v8f acc = {0.f, 0.f, 0.f, 0.f, 0.f, 0.f, 0.f, 0.f};

    for (int it4 = 0; it4 < iters; it4 += 4) {
        // keep 15 concurrent row streams fed (~1KB ahead per row)
        __builtin_prefetch(drow + it4 * 4 + 256, 0, 0);
        #pragma unroll
        for (int u = 0; u < 4; ++u) {
            const int n = (it4 + u) * 4 + kk;
            float2 dd = *(const float2*)(drow + n);
            int2   ll = *(const int2*)(lrow + n);

            float ax = (ll.x == cls) ? __expf(dd.x) : 0.0f;
            float ay = (ll.y == cls) ? __expf(dd.y) : 0.0f;
            float bx = onesCol ? 1.0f : (rowValid ? dd.x : 0.0f);
            float by = onesCol ? 1.0f : (rowValid ? dd.y : 0.0f);

            v2f a; a.x = ax; a.y = ay;
            v2f bf; bf.x = bx; bf.y = by;
            // D = A(16x4,f32) x B(4x16,f32) + C
            acc = __builtin_amdgcn_wmma_f32_16x16x4_f32(
                      false, a, false, bf, (short)0, acc, false, false);
        }
    }

    // D layout: vgpr i, lane L -> (M = i + 8*(L>>4), Ncol = L&15)
    float* Tt = T + ((size_t)b * NTILE + t) * 256;
    const int colN = lane & 15;
    const int mhi  = (lane >> 4) * 8;
    #pragma unroll
    for (int i = 0; i < 8; ++i)
        atomicAdd(&Tt[(mhi + i) * 16 + colN], acc[i]);
}

// ---------------------------------------------------------------------------
// 3) Finalization: 2 scales x 16 b x 20 classes x 10 pairs = 6400 items
// ---------------------------------------------------------------------------
__global__ void kld_final_kernel(const int* __restrict__ counts,
                                 const float* __restrict__ T,
                                 float* __restrict__ out) {
    const int PJ[10] = {0,0,0,0,1,1,1,2,2,3};
    const int PK[10] = {1,2,3,4,2,3,4,3,4,4};
    float tot = 0.0f; int cnt = 0;

    for (int id = threadIdx.x; id < 2 * BB * NCLS * 10; id += 256) {
        int s = id / (BB * NCLS * 10);
        int rem = id - s * (BB * NCLS * 10);
        int b = rem / (NCLS * 10); rem -= b * (NCLS * 10);
        int cls = rem / 10;
        int pi  = rem - cls * 10;

        if (counts[(s * BB + b) * NCLS + cls] < 2) continue;

        int t = cls / 3;                       // 3 classes per 15-row tile
        int base = (cls - t * 3) * 5;
        int rj = base + PJ[pi], rk = base + PK[pi];

        const float* Tt = T + (((size_t)s * BB + b) * NTILE + t) * 256;
        float Zj = Tt[rj * 16 + 15], Zk = Tt[rk * 16 + 15];
        float lsej = logf(Zj), lsek = logf(Zk);
        float Sjj = Tt[rj * 16 + rj] / Zj, Sjk = Tt[rj * 16 + rk] / Zj;
        float Skj = Tt[rk * 16 + rj] / Zk, Skk = Tt[rk * 16 + rk] / Zk;
        float Ej = Sjj - lsej, Ek = Skk - lsek;
        float kld = 0.5f * (Ej + Ek - (Sjk - lsek) - (Skj - lsej));
        tot += expf(-kld);
        cnt += 1;
    }

    __shared__ float st[256];
    __shared__ int   sc[256];
    st[threadIdx.x] = tot; sc[threadIdx.x] = cnt;
    __syncthreads();
    for (int o = 128; o > 0; o >>= 1) {
        if (threadIdx.x < o) {
            st[threadIdx.x] += st[threadIdx.x + o];
            sc[threadIdx.x] += sc[threadIdx.x + o];
        }
        __syncthreads();
    }
    if (threadIdx.x == 0)
        out[0] = (sc[0] > 0) ? (st[0] / (float)sc[0]) : 0.0f;
}

// ---------------------------------------------------------------------------
// launch
// ---------------------------------------------------------------------------
extern "C" void kernel_launch(void* const* d_in, const int* in_sizes, int n_in,
                              void* d_out, int out_size, void* d_ws, size_t ws_size,
                              hipStream_t stream) {
    const float* dist0    = (const float*)d_in[0];   // [16,100,128,128]
    const float* dist1    = (const float*)d_in[1];   // [16,100,64,64]
    const int*   labels   = (const int*)d_in[2];     // [16,512,512]
    const int*   protoCls = (const int*)d_in[3];     // [200]
    float*       out      = (float*)d_out;

    // workspace layout
    int* labFlat0 = (int*)d_ws;                        // 16*16384
    int* labFlat1 = labFlat0 + BB * 16384;             // 16*4096
    int* counts   = labFlat1 + BB * 4096;              // 2*16*20 (padded 1024)
    float* T      = (float*)(counts + 1024);           // 2*16*7*256 floats

    const int zeroWords = 1024 + 2 * BB * NTILE * 256; // counts pad + T
    kld_zero_kernel<<<(zeroWords + 255) / 256, 256, 0, stream>>>(counts, zeroWords);

    kld_labels_kernel<<<dim3(BB, 8, 2), 256, 0, stream>>>(labels, labFlat0, labFlat1, counts);

    // scale 0: N=16384, 4 pixel partitions; scale 1: N=4096, 2 partitions
    kld_accum_kernel<16384, 4><<<dim3(4, NTILE, BB), 256, 0, stream>>>(
        dist0, labFlat0, protoCls, T);
    kld_accum_kernel<4096, 2><<<dim3(2, NTILE, BB), 256, 0, stream>>>(
        dist1, labFlat1, protoCls + 100, T + (size_t)BB * NTILE * 256);

    kld_final_kernel<<<1, 256, 0, stream>>>(counts, T, out);
}